// GraphCNN_71536975282752
// MI455X (gfx1250) — compile-verified
//
#include <hip/hip_runtime.h>

typedef float v2f __attribute__((ext_vector_type(2)));
typedef float v8f __attribute__((ext_vector_type(8)));
typedef unsigned int v4u __attribute__((ext_vector_type(4)));
typedef int v8i __attribute__((ext_vector_type(8)));
typedef int v4i __attribute__((ext_vector_type(4)));

#define B_ 32
#define P_ 1024
#define F_ 512
#define C_ 512

// GEMM tiling: block 128(M) x 64(N), K-stage 32, 8 waves (wave32),
// wave grid 4(M) x 2(N): each wave owns 32x32 -> 4 f32 WMMA accumulators.
#define TMB 128
#define TNB 64
#define TK 32

// LDS row strides (floats). Chosen so the two half-wave rows of a fragment
// land in disjoint bank halves: 2*RS mod 64 == 16.
#define ARS1 136 /* gft A:  [TK][128+8]  pad 8 DW per 128 DW row */
#define ARS2 34  /* xupd A: [128][32+2]  pad 2 DW per 32 DW row  */
#define BRS 72   /* B:      [TK][64+8]   pad 8 DW per 64 DW row  */
#define ASZ1 (TK * ARS1)
#define ASZ2 (TMB * ARS2)
#define BSZ (TK * BRS)

// ---------------------------------------------------------------------------
// Tensor Data Mover: 2D tile load, global -> LDS, with hardware LDS padding.
// D# layout per CDNA5 ISA (group0 128b, group1 256b).
// pad codes: interval 0=2DW,1=4,...,7=256 ; amount 0=1DW,...,127=128DW
// This toolchain's builtin is the 6-arg variant (clang-23 / therock headers).
// ---------------------------------------------------------------------------
__device__ __forceinline__ void tdm_load_2d(unsigned lds_addr, const float* gaddr,
                                            int tile_w, int tile_h, int row_stride,
                                            int pad_interval_code, int pad_amount_code) {
    unsigned long long ga = (unsigned long long)(const void*)gaddr;
    v4u g0;
    g0[0] = 1u;                                  // count=1, user mode, no gather
    g0[1] = lds_addr;                            // LDS byte address
    g0[2] = (unsigned)(ga & 0xFFFFFFFFu);        // global_addr[31:0]
    g0[3] = (unsigned)((ga >> 32) & 0x01FFFFFFu) // global_addr[56:32]
            | (2u << 30);                        // type=2 ("image")
    v8i g1;
    g1[0] = (2 << 16)                            // data_size = 4 bytes
            | (1 << 20)                          // pad_enable (load only)
            | (pad_interval_code << 22)
            | (pad_amount_code << 25);           // workgroup_mask=0 (no cluster)
    g1[1] = (row_stride & 0xFFFF) << 16;         // tensor_dim0[15:0]  (bits 63:48)
    g1[2] = ((row_stride >> 16) & 0xFFFF)        // tensor_dim0[31:16]
            | ((1 << 20) & 0xFFFF0000);          // tensor_dim1 low16 = 16 (bit 20 set here)
    g1[3] = ((1 << 20) >> 16)                    // tensor_dim1 high16 = 16 (dim1 = 2^20 + 16, no clip)
            | ((tile_w & 0xFFFF) << 16);         // tile_dim0
    g1[4] = (tile_h & 0xFFFF);                   // tile_dim1 ; tile_dim2 = 0
    g1[5] = row_stride;                          // tensor_dim0_stride[31:0]
    g1[6] = 0;                                   // stride hi + dim1_stride lo
    g1[7] = 0;
    v4i z4 = {0, 0, 0, 0};                       // groups 2/3 unused for 2D tiles
    v8i z8 = {0, 0, 0, 0, 0, 0, 0, 0};
    __builtin_amdgcn_tensor_load_to_lds(g0, g1, z4, z4, z8, 0);
}

// ---------------------------------------------------------------------------
// out0 = x ; out2[b,k,:,:] = x[b] for k=0..2   (float4 grid-stride copy)
// ---------------------------------------------------------------------------
__global__ __launch_bounds__(256) void bcast_kernel(const float* __restrict__ x,
                                                    float* __restrict__ out0,
                                                    float* __restrict__ out2) {
    const size_t PF4 = (size_t)P_ * F_ / 4;
    const size_t total = (size_t)B_ * PF4;
    const float4* x4 = (const float4*)x;
    float4* o0 = (float4*)out0;
    float4* o2 = (float4*)out2;
    for (size_t i = (size_t)blockIdx.x * 256u + threadIdx.x; i < total;
         i += (size_t)gridDim.x * 256u) {
        float4 v = x4[i];
        o0[i] = v;
        size_t b = i / PF4, r = i - b * PF4;
        size_t base = b * 3 * PF4 + r;
        o2[base] = v;
        o2[base + PF4] = v;
        o2[base + 2 * PF4] = v;
    }
}

// ---------------------------------------------------------------------------
// updated_evectors: out3[b,c,p] = E[b, p, round(c*(N-1)/(C-1))]
// ---------------------------------------------------------------------------
__global__ __launch_bounds__(256) void evsel_kernel(const float* __restrict__ E,
                                                    const int* __restrict__ nv,
                                                    float* __restrict__ out3) {
    // grid: (P_/64, C_/32, B_)
    const int b = blockIdx.z;
    const int c0 = blockIdx.y * 32;
    const int p0 = blockIdx.x * 64;
    const int N = nv[b];
    const float step = ((float)N - 1.0f) / (float)(C_ - 1);

    __shared__ float tile[32][64 + 1];
    __shared__ int idxs[32];

    const int t = threadIdx.x;
    if (t < 32) {
        int id = (int)rintf((float)(c0 + t) * step);  // round-half-even == jnp.round
        id = id < 0 ? 0 : (id > P_ - 1 ? P_ - 1 : id);
        idxs[t] = id;
    }
    __syncthreads();

    const float* Eb = E + (size_t)b * P_ * P_;
    {
        int cl = t & 31, po = t >> 5;
        int id = idxs[cl];
#pragma unroll
        for (int pp = 0; pp < 64; pp += 8)
            tile[cl][po + pp] = Eb[(size_t)(p0 + po + pp) * P_ + id];
    }
    __syncthreads();
    float* O = out3 + (size_t)b * C_ * P_;
    {
        int pl = t & 63, co = t >> 6;
#pragma unroll
        for (int cc = 0; cc < 32; cc += 4)
            O[(size_t)(c0 + co + cc) * P_ + (p0 + pl)] = tile[co + cc][pl];
    }
}

// ---------------------------------------------------------------------------
// GEMM 1: gft[b] = E[b]^T @ X[b].  A[m=p][k=n] = E[n][p]: E rows are K-major,
// so both tiles are plain 2D TDM loads.  Double-buffered TDM pipeline.
// ---------------------------------------------------------------------------
__global__ __launch_bounds__(256) void gft_kernel(const float* __restrict__ E,
                                                  const float* __restrict__ X,
                                                  const int* __restrict__ nv,
                                                  float* __restrict__ gft) {
    // grid: (F_/TNB, P_/TMB, B_)
    const int b = blockIdx.z;
    const int p0 = blockIdx.y * TMB;
    const int f0 = blockIdx.x * TNB;
    const int N = nv[b];

    __shared__ float As[2 * ASZ1];  // [buf][TK][ARS1] : As[k][m] = E[n0+k][p0+m]
    __shared__ float Bs[2 * BSZ];   // [buf][TK][BRS]  : Bs[k][f] = X[n0+k][f0+f]

    const int t = threadIdx.x;
    const int lane = t & 31;
    const int w = t >> 5;
    const int wm = w >> 1;       // 0..3
    const int wn = w & 1;        // 0..1
    const int half = lane >> 4;  // f32 WMMA k-pair select
    const int l16 = lane & 15;

    const float* Eb = E + (size_t)b * P_ * P_;
    const float* Xb = X + (size_t)b * P_ * F_;

    v8f acc[2][2] = {};

    const unsigned ldsA = (unsigned)(size_t)(void*)As;
    const unsigned ldsB = (unsigned)(size_t)(void*)Bs;

    const int nstages = (N + TK - 1) / TK;  // rows >= N are exactly zero: skip

    if (w == 0) {
        tdm_load_2d(ldsA, Eb + p0, TMB, TK, P_, /*128DW*/ 6, /*+8DW*/ 7);
        tdm_load_2d(ldsB, Xb + f0, TNB, TK, F_, /*64DW*/ 5, /*+8DW*/ 7);
    }
    int buf = 0;
    for (int s = 0; s < nstages; ++s) {
        if (w == 0) __builtin_amdgcn_s_wait_tensorcnt(0);
        __syncthreads();  // current buffer ready for all waves
        if (s + 1 < nstages && w == 0) {  // prefetch next stage into other buffer
            const int n0 = (s + 1) * TK;
            tdm_load_2d(ldsA + (unsigned)((buf ^ 1) * ASZ1 * 4),
                        Eb + (size_t)n0 * P_ + p0, TMB, TK, P_, 6, 7);
            tdm_load_2d(ldsB + (unsigned)((buf ^ 1) * BSZ * 4),
                        Xb + (size_t)n0 * F_ + f0, TNB, TK, F_, 5, 7);
        }
        const float* A = As + buf * ASZ1;
        const float* Bb = Bs + buf * BSZ;
#pragma unroll
        for (int k = 0; k < TK; k += 4) {
            const int kk = k + 2 * half;
            v2f a[2], bb[2];
#pragma unroll
            for (int i = 0; i < 2; ++i) {
                const int m = wm * 32 + i * 16 + l16;
                a[i].x = A[kk * ARS1 + m];
                a[i].y = A[(kk + 1) * ARS1 + m];
            }
#pragma unroll
            for (int j = 0; j < 2; ++j) {
                const int n = wn * 32 + j * 16 + l16;
                bb[j].x = Bb[kk * BRS + n];
                bb[j].y = Bb[(kk + 1) * BRS + n];
            }
#pragma unroll
            for (int i = 0; i < 2; ++i)
#pragma unroll
                for (int j = 0; j < 2; ++j)
                    acc[i][j] = __builtin_amdgcn_wmma_f32_16x16x4_f32(
                        false, a[i], false, bb[j], (short)0, acc[i][j], false, false);
        }
        buf ^= 1;
    }

    float* G = gft + (size_t)b * P_ * F_;
#pragma unroll
    for (int i = 0; i < 2; ++i) {
        const int mrow = p0 + wm * 32 + i * 16 + 8 * half;
#pragma unroll
        for (int j = 0; j < 2; ++j) {
            const int ncol = f0 + wn * 32 + j * 16 + l16;
#pragma unroll
            for (int v = 0; v < 8; ++v)
                G[(size_t)(mrow + v) * F_ + ncol] = acc[i][j][v];
        }
    }
}

// ---------------------------------------------------------------------------
// GEMM 2: x_updated[b,ks] = E[b] @ (mask_ks * gft[b]).
// A tile via TDM (M-major -> single b64 fragment loads); B tile manually
// zero-masked to the split's [lo,hi) row range, software-prefetched.
// K-loop covers only tiles intersecting [lo,hi): 3 splits == 1 full GEMM.
// ---------------------------------------------------------------------------
__global__ __launch_bounds__(256) void xupd_kernel(const float* __restrict__ E,
                                                   const float* __restrict__ G,
                                                   const int* __restrict__ nv,
                                                   float* __restrict__ out1) {
    // grid: (F_/TNB, (P_/TMB)*3, B_)
    const int b = blockIdx.z;
    const int ks = blockIdx.y / (P_ / TMB);
    const int n0 = (blockIdx.y % (P_ / TMB)) * TMB;
    const int f0 = blockIdx.x * TNB;
    const int N = nv[b];

    // replicate JAX f32 semantics: ceil(N*0.1f), ceil(N*0.1f*2.0f)
    const float nf = (float)N;
    const int s1 = (int)ceilf(nf * 0.1f);
    const int s2 = (int)ceilf(nf * 0.1f * 2.0f);
    int lo, hi;
    if (ks == 0)      { lo = 0;       hi = s1; }
    else if (ks == 1) { lo = s1;      hi = s1 + s2; }
    else              { lo = s1 + s2; hi = N; }

    __shared__ float As[2 * ASZ2];  // [buf][TMB][ARS2] : As[m][k] = E[n0+m][kt+k]
    __shared__ float Bs[2 * BSZ];   // [buf][TK][BRS]   : masked gft rows

    const int t = threadIdx.x;
    const int lane = t & 31;
    const int w = t >> 5;
    const int wm = w >> 1;
    const int wn = w & 1;
    const int half = lane >> 4;
    const int l16 = lane & 15;

    const float* Eb = E + (size_t)b * P_ * P_;
    const float* Gb = G + (size_t)b * P_ * F_;

    v8f acc[2][2] = {};

    const unsigned ldsA = (unsigned)(size_t)(void*)As;

    const int klo = lo & ~(TK - 1);
    const int nstages = (hi - klo + TK - 1) / TK;

    const int lf = t & 63, lk4 = t >> 6;  // manual B loader mapping

    // prologue: stage 0
    if (w == 0)
        tdm_load_2d(ldsA, Eb + (size_t)n0 * P_ + klo, TK, TMB, P_, /*32DW*/ 4, /*+2DW*/ 1);
    {
        float* Bw = Bs;
#pragma unroll
        for (int kk = 0; kk < TK; kk += 4) {
            const int p = klo + lk4 + kk;
            float v = 0.0f;
            if (p >= lo && p < hi) v = Gb[(size_t)p * F_ + (f0 + lf)];
            Bw[(lk4 + kk) * BRS + lf] = v;
        }
    }

    int buf = 0;
    for (int s = 0; s < nstages; ++s) {
        if (w == 0) __builtin_amdgcn_s_wait_tensorcnt(0);
        __syncthreads();
        if (s + 1 < nstages) {  // prefetch next stage
            const int kt = klo + (s + 1) * TK;
            if (w == 0)
                tdm_load_2d(ldsA + (unsigned)((buf ^ 1) * ASZ2 * 4),
                            Eb + (size_t)n0 * P_ + kt, TK, TMB, P_, 4, 1);
            float* Bw = Bs + (buf ^ 1) * BSZ;
#pragma unroll
            for (int kk = 0; kk < TK; kk += 4) {
                const int p = kt + lk4 + kk;
                float v = 0.0f;
                if (p >= lo && p < hi) v = Gb[(size_t)p * F_ + (f0 + lf)];
                Bw[(lk4 + kk) * BRS + lf] = v;
            }
        }
        const float* A = As + buf * ASZ2;
        const float* Bb = Bs + buf * BSZ;
#pragma unroll
        for (int k = 0; k < TK; k += 4) {
            const int kk = k + 2 * half;
            v2f a[2], bb[2];
#pragma unroll
            for (int i = 0; i < 2; ++i) {
                const int m = wm * 32 + i * 16 + l16;
                a[i].x = A[m * ARS2 + kk];      // contiguous pair -> ds_load_b64
                a[i].y = A[m * ARS2 + kk + 1];
            }
#pragma unroll
            for (int j = 0; j < 2; ++j) {
                const int n = wn * 32 + j * 16 + l16;
                bb[j].x = Bb[kk * BRS + n];
                bb[j].y = Bb[(kk + 1) * BRS + n];
            }
#pragma unroll
            for (int i = 0; i < 2; ++i)
#pragma unroll
                for (int j = 0; j < 2; ++j)
                    acc[i][j] = __builtin_amdgcn_wmma_f32_16x16x4_f32(
                        false, a[i], false, bb[j], (short)0, acc[i][j], false, false);
        }
        buf ^= 1;
    }

    float* O = out1 + ((size_t)(b * 3 + ks) * P_ + n0) * F_;
#pragma unroll
    for (int i = 0; i < 2; ++i) {
        const int mrow = wm * 32 + i * 16 + 8 * half;
#pragma unroll
        for (int j = 0; j < 2; ++j) {
            const int ncol = f0 + wn * 32 + j * 16 + l16;
#pragma unroll
            for (int v = 0; v < 8; ++v)
                O[(size_t)(mrow + v) * F_ + ncol] = acc[i][j][v];
        }
    }
}

// ---------------------------------------------------------------------------
extern "C" void kernel_launch(void* const* d_in, const int* in_sizes, int n_in,
                              void* d_out, int out_size, void* d_ws, size_t ws_size,
                              hipStream_t stream) {
    (void)in_sizes; (void)n_in; (void)out_size; (void)ws_size;

    const float* x  = (const float*)d_in[0];  // [B,P,F]
    const float* ev = (const float*)d_in[1];  // [B,P,P]
    const int*   nv = (const int*)d_in[2];    // [B]

    float* out = (float*)d_out;
    const size_t PF = (size_t)P_ * F_;
    float* out0 = out;                         // n_level_feat  [B,P,F]
    float* out1 = out0 + (size_t)B_ * PF;      // x_updated     [B,3,P,F]
    float* out2 = out1 + (size_t)B_ * 3 * PF;  // original_x    [B,3,P,F]
    float* out3 = out2 + (size_t)B_ * 3 * PF;  // updated_evec  [B,C,P]

    float* gft = (float*)d_ws;                 // scratch [B,P,F] = 64 MB

    dim3 blk(256);
    bcast_kernel<<<2048, blk, 0, stream>>>(x, out0, out2);
    evsel_kernel<<<dim3(P_ / 64, C_ / 32, B_), blk, 0, stream>>>(ev, nv, out3);
    gft_kernel<<<dim3(F_ / TNB, P_ / TMB, B_), blk, 0, stream>>>(ev, x, nv, gft);
    xupd_kernel<<<dim3(F_ / TNB, (P_ / TMB) * 3, B_), blk, 0, stream>>>(ev, gft, nv, out1);
}